// MGCNLayerWrapper_11931419148745
// MI455X (gfx1250) — compile-verified
//
#include <hip/hip_runtime.h>
#include <hip/hip_bf16.h>
#include <math.h>

#define D 128

typedef float v2f __attribute__((ext_vector_type(2)));
typedef float v8f __attribute__((ext_vector_type(8)));
typedef unsigned int u32x4 __attribute__((ext_vector_type(4)));
typedef int i32x8 __attribute__((ext_vector_type(8)));
typedef int i32x4 __attribute__((ext_vector_type(4)));

// ---------------------------------------------------------------------------
// Zero a float buffer.
__global__ __launch_bounds__(256)
void zero_kernel(float* __restrict__ p, int count) {
    int i = blockIdx.x * blockDim.x + threadIdx.x;
    if (i < count) p[i] = 0.0f;
}

// ---------------------------------------------------------------------------
// Relational edge scatter: one wave32 per edge.
// msg = x[src] * rel_emb[type]; atomic add into agg[dst]; optional degree.
__global__ __launch_bounds__(256)
void scatter_edges(const float* __restrict__ x, const float* __restrict__ rel,
                   const int* __restrict__ ei,  const int* __restrict__ et,
                   float* __restrict__ agg, float* __restrict__ deg,
                   int E, int add_deg) {
    int wid  = (blockIdx.x * blockDim.x + threadIdx.x) >> 5;
    int lane = threadIdx.x & 31;
    if (wid >= E) return;
    int src = ei[wid];
    int dst = ei[E + wid];
    int typ = et[wid];
    const float4 xv = ((const float4*)(x   + (size_t)src * D))[lane];
    const float4 rv = ((const float4*)(rel + (size_t)typ * D))[lane];
    float* a = agg + (size_t)dst * D + lane * 4;
    unsafeAtomicAdd(a + 0, xv.x * rv.x);
    unsafeAtomicAdd(a + 1, xv.y * rv.y);
    unsafeAtomicAdd(a + 2, xv.z * rv.z);
    unsafeAtomicAdd(a + 3, xv.w * rv.w);
    if (add_deg && lane == 0) unsafeAtomicAdd(deg + dst, 1.0f);
}

// ---------------------------------------------------------------------------
// Jump diffusion scatter: one wave32 per jump edge.
// dch[dst] += jump_weight * edge_w[e] * emb[src]
__global__ __launch_bounds__(256)
void scatter_jump(const float* __restrict__ emb, const float* __restrict__ ew,
                  const float* __restrict__ jw, const int* __restrict__ eij,
                  float* __restrict__ dch, int EJ) {
    int wid  = (blockIdx.x * blockDim.x + threadIdx.x) >> 5;
    int lane = threadIdx.x & 31;
    if (wid >= EJ) return;
    int src = eij[wid];
    int dst = eij[EJ + wid];
    float w = jw[0] * ew[wid];
    const float4 xv = ((const float4*)(emb + (size_t)src * D))[lane];
    float* a = dch + (size_t)dst * D + lane * 4;
    unsafeAtomicAdd(a + 0, w * xv.x);
    unsafeAtomicAdd(a + 1, w * xv.y);
    unsafeAtomicAdd(a + 2, w * xv.z);
    unsafeAtomicAdd(a + 3, w * xv.w);
}

// ---------------------------------------------------------------------------
// TDM: DMA a 128x128 fp32 matrix (row-major, stride 128) from global to LDS.
// D# per CDNA5 ISA ch.8: group0 = {flags, lds_addr, global_addr, type},
// group1 = {data_size, tensor dims, tile dims, strides}; groups 2/3 zero (2D).
// 6-arg builtin form (amdgpu-toolchain / clang-23).
__device__ __forceinline__ void tdm_load_w(const float* gsrc, void* lds_dst) {
    const uint64_t ga = (uint64_t)(uintptr_t)gsrc;
    const uint32_t la = (uint32_t)(uintptr_t)lds_dst;   // LDS aperture: addr[31:0]
    u32x4 g0;
    g0[0] = 1u;                                          // count=1 valid descriptor
    g0[1] = la;                                          // lds_addr
    g0[2] = (uint32_t)(ga & 0xFFFFFFFFu);                // global_addr[31:0]
    g0[3] = (uint32_t)((ga >> 32) & 0x1FFFFFFu)          // global_addr[56:32]
          | (2u << 30);                                  // type = 2 ("image")
    i32x8 g1;
    g1[0] = (int)(2u << 16);       // data_size=2 (4 bytes); mask/flags = 0
    g1[1] = (int)(128u << 16);     // tensor_dim0[15:0] @ bits 63:48
    g1[2] = (int)(128u << 16);     // tensor_dim0 hi=0 | tensor_dim1[15:0]
    g1[3] = (int)(128u << 16);     // tensor_dim1 hi=0 | tile_dim0=128
    g1[4] = 128;                   // tile_dim1=128 | tile_dim2=0
    g1[5] = 128;                   // tensor_dim0_stride[31:0]
    g1[6] = 0;                     // stride hi | tensor_dim1_stride lo
    g1[7] = 0;
    const i32x4 z4 = {0, 0, 0, 0};
    const i32x8 z8 = {0, 0, 0, 0, 0, 0, 0, 0};
    __builtin_amdgcn_tensor_load_to_lds(g0, g1, z4, z4, z8, 0);
}

// ---------------------------------------------------------------------------
// Fused dense step: out = x + res * tanh( (agg/deg) @ W + x @ Wloop )
// Block = 256 threads = 8 waves; block owns 16 rows, wave w owns N-tile w.
// Both 64KB weight matrices are DMA'd into double-buffered LDS by the TDM;
// s_wait_tensorcnt(1) gates the first K-loop, (0) gates the second, so the
// Wloop DMA overlaps the entire first K-loop.
__global__ __launch_bounds__(256)
void gemm_fused(const float* __restrict__ agg, const float* __restrict__ deg,
                const float* __restrict__ x,
                const float* __restrict__ W, const float* __restrict__ Wl,
                const float* __restrict__ res, float* __restrict__ out) {
    __shared__ float sB0[D * D];          // 64 KB: W
    __shared__ float sB1[D * D];          // 64 KB: Wloop

    const int tid  = threadIdx.x;
    const int lane = tid & 31;
    const int wave = tid >> 5;            // N-tile index, 0..7
    const int row0 = blockIdx.x * 16;

    const bool issuer = (tid == 0);       // wave 0 issues both DMAs
    if (issuer) {
        tdm_load_w(W,  sB0);              // TENSORcnt -> 1
        tdm_load_w(Wl, sB1);              // TENSORcnt -> 2
        __builtin_amdgcn_s_wait_tensorcnt(1);   // W landed (in-order)
    }
    __syncthreads();                      // publish sB0 to all waves

    const int m    = lane & 15;           // row-in-tile (A), col-in-tile (B)
    const int kb   = (lane >> 4) * 2;     // K sub-offset: 0 / 2
    const int row  = row0 + m;
    const int ncol = wave * 16 + m;

    const float inv = 1.0f / fmaxf(deg[row], 1.0f);
    const float* __restrict__ arow = agg + (size_t)row * D;
    const float* __restrict__ xrow = x   + (size_t)row * D;

    v8f c = {0.f, 0.f, 0.f, 0.f, 0.f, 0.f, 0.f, 0.f};

    // (agg * inv) @ W
    #pragma unroll 4
    for (int k = 0; k < D; k += 4) {
        v2f a, b;
        a.x = arow[k + kb]     * inv;
        a.y = arow[k + kb + 1] * inv;
        b.x = sB0[(k + kb)     * D + ncol];
        b.y = sB0[(k + kb + 1) * D + ncol];
        c = __builtin_amdgcn_wmma_f32_16x16x4_f32(false, a, false, b,
                                                  (short)0, c, false, false);
    }

    if (issuer) __builtin_amdgcn_s_wait_tensorcnt(0);  // Wloop landed
    __syncthreads();                      // publish sB1

    // + x @ Wloop
    #pragma unroll 4
    for (int k = 0; k < D; k += 4) {
        v2f a, b;
        a.x = xrow[k + kb];
        a.y = xrow[k + kb + 1];
        b.x = sB1[(k + kb)     * D + ncol];
        b.y = sB1[(k + kb + 1) * D + ncol];
        c = __builtin_amdgcn_wmma_f32_16x16x4_f32(false, a, false, b,
                                                  (short)0, c, false, false);
    }

    // C/D layout: VGPR i, lanes 0-15 -> M=i, N=lane; lanes 16-31 -> M=i+8.
    const float r     = res[0];
    const int   nout  = wave * 16 + m;
    const int   mbase = row0 + (lane >> 4) * 8;
    #pragma unroll
    for (int i = 0; i < 8; ++i) {
        const size_t idx = (size_t)(mbase + i) * D + nout;
        out[idx] = x[idx] + r * tanhf(c[i]);
    }
}

// ---------------------------------------------------------------------------
extern "C" void kernel_launch(void* const* d_in, const int* in_sizes, int n_in,
                              void* d_out, int out_size, void* d_ws, size_t ws_size,
                              hipStream_t stream) {
    const float* emb    = (const float*)d_in[0];
    const float* change = (const float*)d_in[1];
    const float* W1     = (const float*)d_in[2];
    const float* Wl1    = (const float*)d_in[3];
    const float* rel1   = (const float*)d_in[4];
    const float* W2     = (const float*)d_in[5];
    const float* Wl2    = (const float*)d_in[6];
    const float* rel2   = (const float*)d_in[7];
    const float* res    = (const float*)d_in[8];
    const float* jw     = (const float*)d_in[9];
    const float* ewj    = (const float*)d_in[10];
    const int*   ei     = (const int*)d_in[11];
    const int*   et     = (const int*)d_in[12];
    const int*   eij    = (const int*)d_in[13];

    const int N  = in_sizes[0] / D;       // 100000
    const int E  = in_sizes[11] / 2;      // 600000
    const int EJ = in_sizes[10];          // 300000

    float* agg = (float*)d_ws;            // N*D
    float* deg = agg + (size_t)N * D;     // N
    float* h   = deg + N;                 // N*D

    float* out_change = (float*)d_out;
    float* dch        = out_change + (size_t)N * D;

    // change passthrough
    (void)hipMemcpyAsync(out_change, change, (size_t)N * D * sizeof(float),
                         hipMemcpyDeviceToDevice, stream);

    const int zc1 = N * D + N;
    zero_kernel<<<(zc1 + 255) / 256, 256, 0, stream>>>(agg, zc1);

    const int eb  = (E * 32 + 255) / 256;
    const int ejb = (EJ * 32 + 255) / 256;
    const int gb  = N / 16;               // 6250, exact

    // Layer 1
    scatter_edges<<<eb, 256, 0, stream>>>(emb, rel1, ei, et, agg, deg, E, 1);
    gemm_fused<<<gb, 256, 0, stream>>>(agg, deg, emb, W1, Wl1, res, h);

    // Layer 2 (degree identical: same dst list)
    zero_kernel<<<(N * D + 255) / 256, 256, 0, stream>>>(agg, N * D);
    scatter_edges<<<eb, 256, 0, stream>>>(h, rel2, ei, et, agg, deg, E, 0);
    gemm_fused<<<gb, 256, 0, stream>>>(agg, deg, h, W2, Wl2, res, dch);

    // Jump diffusion accumulates on top of dch
    scatter_jump<<<ejb, 256, 0, stream>>>(emb, ewj, jw, eij, dch, EJ);
}